// GNNClassifier_15831249453219
// MI455X (gfx1250) — compile-verified
//
#include <hip/hip_runtime.h>
#include <math.h>

#define N_NODES 100000
#define N_EDGES 800000
#define IN_DIM  128
#define HID_DIM 1024
#define OUT_DIM 40
#define TOT_EDGES (N_EDGES + N_NODES)

typedef __attribute__((ext_vector_type(2))) float v2f;
typedef __attribute__((ext_vector_type(8))) float v8f;

// ---------------- degree + normalization ----------------
__global__ __launch_bounds__(256) void degree_kernel(const long long* __restrict__ dst,
                                                     float* __restrict__ deg) {
    int e = blockIdx.x * 256 + threadIdx.x;
    if (e < N_EDGES) {
        long long d = dst[e];
        unsafeAtomicAdd(&deg[d], 1.0f);   // -> global_atomic_add_f32
    }
}

__global__ __launch_bounds__(256) void rsqrt_kernel(float* __restrict__ deg) {
    int i = blockIdx.x * 256 + threadIdx.x;
    if (i < N_NODES) {
        // +1 accounts for the self-loop; deg >= 1 so rsqrt is safe
        deg[i] = rsqrtf(deg[i] + 1.0f);
    }
}

// ---------------- layer-1 aggregation in 128-dim ----------------
// One wave32 per (edge|self-loop); lane owns a contiguous float4 of the feature.
__global__ __launch_bounds__(32) void agg1_kernel(const long long* __restrict__ src,
                                                  const long long* __restrict__ dst,
                                                  const float* __restrict__ dis,
                                                  const float* __restrict__ x,
                                                  float* __restrict__ xa) {
    int e = blockIdx.x;
    int lane = threadIdx.x;
    long long s, t;
    if (e < N_EDGES) { s = src[e]; t = dst[e]; }
    else             { s = t = (long long)(e - N_EDGES); }
    float norm = dis[s] * dis[t];

    const float4* xin = (const float4*)(x + (size_t)s * IN_DIM) + lane;  // b128 load
    float4 v = *xin;
    float* xo = xa + (size_t)t * IN_DIM + lane * 4;
    unsafeAtomicAdd(xo + 0, v.x * norm);
    unsafeAtomicAdd(xo + 1, v.y * norm);
    unsafeAtomicAdd(xo + 2, v.z * norm);
    unsafeAtomicAdd(xo + 3, v.w * norm);
}

// ---------------- GEMM1: xa[N,128] @ W1[128,1024] + b1, ReLU -> h ----------------
// grid (N/16, 1024/128), block 256 = 8 waves; each wave owns a 16x16 tile.
__global__ __launch_bounds__(256) void gemm1_kernel(const float* __restrict__ xa,
                                                    const float* __restrict__ W1,
                                                    const float* __restrict__ b1,
                                                    float* __restrict__ h) {
    __shared__ float As[16][132];  // 528 B pitch: 16B-aligned rows, conflict-free cols
    const int mTile = blockIdx.x;
    const int nBase = blockIdx.y * 128;
    const int tid   = threadIdx.x;

    // stage A tile (16x128) as float4: global_load_b128 + ds_store_b128
    for (int i = tid; i < 16 * 32; i += 256) {
        int r = i >> 5, c4 = i & 31;
        float4 v = *((const float4*)(xa + (size_t)(mTile * 16 + r) * IN_DIM) + c4);
        *(float4*)&As[r][c4 * 4] = v;
    }
    __syncthreads();

    const int wave  = tid >> 5;
    const int lane  = tid & 31;
    const int row16 = lane & 15;
    const int half  = lane >> 4;            // 0: K pair {0,1}, 1: K pair {2,3}
    const int col   = nBase + wave * 16 + row16;

    v8f acc = {};
    for (int k = 0; k < IN_DIM; k += 4) {
        int ka = k + half * 2;
        v2f a, b;
        a.x = As[row16][ka];
        a.y = As[row16][ka + 1];
        b.x = W1[(size_t)ka * HID_DIM + col];
        b.y = W1[(size_t)(ka + 1) * HID_DIM + col];
        acc = __builtin_amdgcn_wmma_f32_16x16x4_f32(false, a, false, b,
                                                    (short)0, acc, false, false);
    }

    const float bias = b1[col];
    const int rowBase = mTile * 16 + half * 8;
    for (int v = 0; v < 8; ++v) {
        float val = acc[v] + bias;
        val = val > 0.0f ? val : 0.0f;      // ReLU
        h[(size_t)(rowBase + v) * HID_DIM + col] = val;
    }
}

// ---------------- GEMM2: h[N,1024] @ W2[1024,40] -> t[N,40] ----------------
// grid N/16, block 96 = 3 waves covering 48 cols (40 valid, guarded).
__global__ __launch_bounds__(96) void gemm2_kernel(const float* __restrict__ h,
                                                   const float* __restrict__ W2,
                                                   float* __restrict__ t) {
    __shared__ float As[16][132];
    const int mTile = blockIdx.x;
    const int tid   = threadIdx.x;
    const int wave  = tid >> 5;
    const int lane  = tid & 31;
    const int row16 = lane & 15;
    const int half  = lane >> 4;
    const int col   = wave * 16 + row16;    // 0..47
    const bool colOk = (col < OUT_DIM);

    v8f acc = {};
    for (int kc = 0; kc < HID_DIM; kc += 128) {
        __syncthreads();
        for (int i = tid; i < 16 * 32; i += 96) {
            int r = i >> 5, c4 = i & 31;
            float4 v = *((const float4*)(h + (size_t)(mTile * 16 + r) * HID_DIM + kc) + c4);
            *(float4*)&As[r][c4 * 4] = v;
        }
        __syncthreads();

        for (int k = 0; k < 128; k += 4) {
            int ka = k + half * 2;
            int gk = kc + ka;
            v2f a, b;
            a.x = As[row16][ka];
            a.y = As[row16][ka + 1];
            b.x = colOk ? W2[(size_t)gk * OUT_DIM + col] : 0.0f;
            b.y = colOk ? W2[(size_t)(gk + 1) * OUT_DIM + col] : 0.0f;
            acc = __builtin_amdgcn_wmma_f32_16x16x4_f32(false, a, false, b,
                                                        (short)0, acc, false, false);
        }
    }

    if (colOk) {
        const int rowBase = mTile * 16 + half * 8;
        for (int v = 0; v < 8; ++v)
            t[(size_t)(rowBase + v) * OUT_DIM + col] = acc[v];
    }
}

// ---------------- layer-2 aggregation in 40-dim ----------------
__global__ __launch_bounds__(64) void agg2_kernel(const long long* __restrict__ src,
                                                  const long long* __restrict__ dst,
                                                  const float* __restrict__ dis,
                                                  const float* __restrict__ tin,
                                                  float* __restrict__ out) {
    int e = blockIdx.x;
    int d = threadIdx.x;
    if (d >= OUT_DIM) return;
    long long s, t;
    if (e < N_EDGES) { s = src[e]; t = dst[e]; }
    else             { s = t = (long long)(e - N_EDGES); }
    float norm = dis[s] * dis[t];
    unsafeAtomicAdd(&out[(size_t)t * OUT_DIM + d], tin[(size_t)s * OUT_DIM + d] * norm);
}

// ---------------- +b2 then row-wise log_softmax (wave32 shuffle reduce) ----------------
__global__ __launch_bounds__(32) void lsm_kernel(float* __restrict__ out,
                                                 const float* __restrict__ b2) {
    const int n    = blockIdx.x;
    const int lane = threadIdx.x;

    float v0 = out[(size_t)n * OUT_DIM + lane] + b2[lane];
    float v1 = -INFINITY;
    if (lane < OUT_DIM - 32)
        v1 = out[(size_t)n * OUT_DIM + lane + 32] + b2[lane + 32];

    float m = fmaxf(v0, v1);
    for (int off = 16; off > 0; off >>= 1)
        m = fmaxf(m, __shfl_xor(m, off, 32));

    float s = __expf(v0 - m) + ((lane < OUT_DIM - 32) ? __expf(v1 - m) : 0.0f);
    for (int off = 16; off > 0; off >>= 1)
        s += __shfl_xor(s, off, 32);

    const float L = m + __logf(s);
    out[(size_t)n * OUT_DIM + lane] = v0 - L;
    if (lane < OUT_DIM - 32)
        out[(size_t)n * OUT_DIM + lane + 32] = v1 - L;
}

// ---------------- launch ----------------
extern "C" void kernel_launch(void* const* d_in, const int* in_sizes, int n_in,
                              void* d_out, int out_size, void* d_ws, size_t ws_size,
                              hipStream_t stream) {
    (void)in_sizes; (void)n_in; (void)out_size; (void)ws_size;

    const float*     x   = (const float*)d_in[0];
    const long long* ei  = (const long long*)d_in[1];   // int64 [2, E]
    const float*     W1  = (const float*)d_in[2];
    const float*     b1  = (const float*)d_in[3];
    const float*     W2  = (const float*)d_in[4];
    const float*     b2  = (const float*)d_in[5];
    const long long* src = ei;
    const long long* dst = ei + N_EDGES;

    char* ws = (char*)d_ws;
    float* dis = (float*)ws;                                   // N floats (deg -> dis in place)
    float* xa  = (float*)(ws + 400128);                        // N*128 floats
    float* t2  = xa;                                           // reuse region for N*40 floats
    float* h   = (float*)(ws + 51600128);                      // N*1024 floats
    float* out = (float*)d_out;

    hipMemsetAsync(dis, 0, (size_t)N_NODES * sizeof(float), stream);
    hipMemsetAsync(xa,  0, (size_t)N_NODES * IN_DIM * sizeof(float), stream);
    hipMemsetAsync(out, 0, (size_t)N_NODES * OUT_DIM * sizeof(float), stream);

    degree_kernel<<<N_EDGES / 256, 256, 0, stream>>>(dst, dis);
    rsqrt_kernel<<<(N_NODES + 255) / 256, 256, 0, stream>>>(dis);
    agg1_kernel<<<TOT_EDGES, 32, 0, stream>>>(src, dst, dis, x, xa);
    gemm1_kernel<<<dim3(N_NODES / 16, HID_DIM / 128), 256, 0, stream>>>(xa, W1, b1, h);
    gemm2_kernel<<<N_NODES / 16, 96, 0, stream>>>(h, W2, t2);
    agg2_kernel<<<TOT_EDGES, 64, 0, stream>>>(src, dst, dis, t2, out);
    lsm_kernel<<<N_NODES, 32, 0, stream>>>(out, b2);
}